// MoELayer_24412594110495
// MI455X (gfx1250) — compile-verified
//
#include <hip/hip_runtime.h>
#include <math.h>

#define N_TOK 4096
#define E_NUM 8
#define C_DIM 1024
#define H_DIM 4096
#define TM    32      // tokens per tile
#define HC    128     // H-chunk per outer iteration
#define KC    128     // K sub-chunk staged in LDS for GEMM1
#define XPAD  8
#define HPAD  4

typedef __attribute__((ext_vector_type(2))) float v2f;
typedef __attribute__((ext_vector_type(8))) float v8f;
typedef int v4i __attribute__((vector_size(16)));   // matches builtin param type

#define AS_GLOBAL __attribute__((address_space(1)))
#define AS_LOCAL  __attribute__((address_space(3)))

#if __has_builtin(__builtin_amdgcn_global_load_async_to_lds_b128)
#define HAS_ASYNC 1
#else
#define HAS_ASYNC 0
#endif

// Copy 16 bytes global -> LDS. Async (ASYNCcnt-tracked) when available.
__device__ __forceinline__ void cp16(const float* g, float* l) {
#if HAS_ASYNC
    __builtin_amdgcn_global_load_async_to_lds_b128(
        (AS_GLOBAL v4i*)g, (AS_LOCAL v4i*)l, 0, 0);
#else
    *(float4*)l = *(const float4*)g;
#endif
}

template <int N> __device__ __forceinline__ void async_wait() {
#if HAS_ASYNC
#if __has_builtin(__builtin_amdgcn_s_wait_asynccnt)
    __builtin_amdgcn_s_wait_asynccnt(N);
#else
    asm volatile("s_wait_asynccnt %0" :: "i"(N) : "memory");
#endif
#endif
}

__device__ __forceinline__ float gelu_exact(float v) {
    return 0.5f * v * (1.0f + erff(v * 0.70710678118654752f));
}

// ---------------------------------------------------------------------------
// Router: one wave32 per token. logits = x . router_w^T, top-2, renormalize,
// atomic-append (token, weight) into per-expert pair lists.
// ---------------------------------------------------------------------------
__global__ __launch_bounds__(256) void moe_router(
    const float* __restrict__ x, const float* __restrict__ rw,
    int* __restrict__ cnt, int* __restrict__ ptok, float* __restrict__ pw)
{
    const int wave = threadIdx.x >> 5;
    const int lane = threadIdx.x & 31;
    const int n = blockIdx.x * 8 + wave;          // grid.x = N_TOK/8 exactly

    const float* xr = x + (size_t)n * C_DIM;
    float logit[E_NUM];
#pragma unroll
    for (int e = 0; e < E_NUM; ++e) {
        const float* wr = rw + (size_t)e * C_DIM;
        float s = 0.f;
        for (int c = lane; c < C_DIM; c += 32)
            s = fmaf(xr[c], wr[c], s);
#pragma unroll
        for (int off = 16; off > 0; off >>= 1)
            s += __shfl_xor(s, off, 32);
        logit[e] = s;
    }
    if (lane == 0) {
        int i0 = 0;
#pragma unroll
        for (int e = 1; e < E_NUM; ++e) if (logit[e] > logit[i0]) i0 = e;
        int i1 = (i0 == 0) ? 1 : 0;
#pragma unroll
        for (int e = 0; e < E_NUM; ++e) {
            if (e == i0) continue;
            if (logit[e] > logit[i1]) i1 = e;
        }
        const float w0 = 1.0f / (1.0f + expf(logit[i1] - logit[i0]));
        const float w1 = 1.0f - w0;
        int p = atomicAdd(&cnt[i0], 1);
        ptok[i0 * N_TOK + p] = n;  pw[i0 * N_TOK + p] = w0;
        p = atomicAdd(&cnt[i1], 1);
        ptok[i1 * N_TOK + p] = n;  pw[i1 * N_TOK + p] = w1;
    }
}

// ---------------------------------------------------------------------------
// Fused expert FFN per (expert, 32-token tile), fp32 WMMA throughout.
//   GEMM1: h = gelu(X W1[e]) chunked over H; W1 K-tiles double-buffered in
//          LDS via async global->LDS copies (ASYNCcnt).
//   GEMM2: out += h W2[e]; per-wave 32x128 fp32 accumulators in VGPRs.
//   Epilogue: weighted global_atomic_add_f32 scatter (2 commutative adds/elem
//   -> bitwise deterministic).
// ---------------------------------------------------------------------------
__global__ __launch_bounds__(256) void moe_expert_gemm(
    const float* __restrict__ x,  const float* __restrict__ w1,
    const float* __restrict__ w2, const int* __restrict__ cnt,
    const int* __restrict__ ptok, const float* __restrict__ pw,
    float* __restrict__ out)
{
    __shared__ float Xs[TM][C_DIM + XPAD];      // 129 KB  (A operand, all chunks)
    __shared__ float W1s[2][KC][HC + HPAD];     // 132 KB  (double-buffered B tile)
    __shared__ float Hs[TM][HC + HPAD];         // ~17 KB  (gelu(h) chunk)
    __shared__ int   tokS[TM];
    __shared__ float wS[TM];

    const int e  = blockIdx.y;
    int nc = cnt[e]; if (nc > N_TOK) nc = N_TOK;
    const int base = blockIdx.x * TM;
    if (base >= nc) return;                     // block-uniform exit

    const int tid = threadIdx.x;
    const int wv  = tid >> 5;                   // wave 0..7
    const int ln  = tid & 31;
    const int lh  = ln >> 4;                    // lane half (0/1)
    const int l16 = ln & 15;
    const int koff = lh * 2;                    // WMMA f32 A/B K-offset per half

    if (tid < TM) {
        const int r = base + tid;
        if (r < nc) { tokS[tid] = ptok[e * N_TOK + r]; wS[tid] = pw[e * N_TOK + r]; }
        else        { tokS[tid] = 0;                   wS[tid] = 0.f; }
    }
    __syncthreads();

    // ---- Stage X tile (32 x 1024): 32 x 16B per thread, async -> LDS ----
#pragma unroll 4
    for (int i = 0; i < 32; ++i) {
        const int idx = i * 256 + tid;
        const int r   = idx >> 8;               // 256 float4 per row
        const int c4  = idx & 255;
        cp16(x + (size_t)tokS[r] * C_DIM + c4 * 4, &Xs[r][c4 * 4]);
    }
    async_wait<0>();
    __syncthreads();

    const float* W1e = w1 + (size_t)e * C_DIM * H_DIM;
    const float* W2e = w2 + (size_t)e * H_DIM * C_DIM;

    v8f oacc[2][8];
#pragma unroll
    for (int mt = 0; mt < 2; ++mt)
#pragma unroll
        for (int nt = 0; nt < 8; ++nt) oacc[mt][nt] = (v8f){};

    for (int hc = 0; hc < H_DIM; hc += HC) {
        if (hc + HC < H_DIM)   // nudge next W2 chunk toward L2
            __builtin_prefetch(W2e + (size_t)(hc + HC) * C_DIM + wv * 128 + l16, 0, 1);

        // ---- GEMM1: wave wv computes h[:, hc + wv*16 .. +16), K = C_DIM ----
        v8f acc0 = (v8f){}, acc1 = (v8f){};

        // prologue: stage W1[k=0..KC, hc..hc+HC) into buffer 0 (16 x 16B/thread)
#pragma unroll 4
        for (int i = 0; i < 16; ++i) {
            const int idx = i * 256 + tid;
            const int r   = idx >> 5;           // 32 float4 per row
            const int c4  = idx & 31;
            cp16(W1e + (size_t)r * H_DIM + hc + c4 * 4, &W1s[0][r][c4 * 4]);
        }

        int buf = 0;
        for (int kk = 0; kk < C_DIM; kk += KC, buf ^= 1) {
            if (kk + KC < C_DIM) {
#pragma unroll 4
                for (int i = 0; i < 16; ++i) {
                    const int idx = i * 256 + tid;
                    const int r   = idx >> 5;
                    const int c4  = idx & 31;
                    cp16(W1e + (size_t)(kk + KC + r) * H_DIM + hc + c4 * 4,
                         &W1s[buf ^ 1][r][c4 * 4]);
                }
                async_wait<16>();               // tile kk complete (in-order)
            } else {
                async_wait<0>();
            }
            __syncthreads();                    // tile visible to all waves

            const int bcol = wv * 16 + l16;
#pragma unroll 4
            for (int k2 = 0; k2 < KC; k2 += 4) {
                v2f b; b.x = W1s[buf][k2 + koff][bcol];
                       b.y = W1s[buf][k2 + koff + 1][bcol];
                v2f a; a.x = Xs[l16][kk + k2 + koff];
                       a.y = Xs[l16][kk + k2 + koff + 1];
                acc0 = __builtin_amdgcn_wmma_f32_16x16x4_f32(
                           false, a, false, b, (short)0, acc0, false, false);
                v2f a2; a2.x = Xs[16 + l16][kk + k2 + koff];
                        a2.y = Xs[16 + l16][kk + k2 + koff + 1];
                acc1 = __builtin_amdgcn_wmma_f32_16x16x4_f32(
                           false, a2, false, b, (short)0, acc1, false, false);
            }
            __syncthreads();                    // safe to overwrite this buffer
        }

        // ---- gelu + share h chunk ----
        const int hcolw = wv * 16 + l16;
#pragma unroll
        for (int j = 0; j < 8; ++j) {           // C/D layout: VGPR j -> M=j / M=8+j
            Hs[j + 8 * lh][hcolw]      = gelu_exact(acc0[j]);
            Hs[16 + j + 8 * lh][hcolw] = gelu_exact(acc1[j]);
        }
        __syncthreads();

        // ---- GEMM2: wave wv owns out cols [wv*128, +128); K = HC ----
        const float* b2base = W2e + (size_t)(hc + koff) * C_DIM + wv * 128 + l16;
#pragma unroll 2
        for (int kh = 0; kh < HC; kh += 4) {
            v2f a0; a0.x = Hs[l16][kh + koff];      a0.y = Hs[l16][kh + koff + 1];
            v2f a1; a1.x = Hs[16 + l16][kh + koff]; a1.y = Hs[16 + l16][kh + koff + 1];
            const float* bp = b2base + (size_t)kh * C_DIM;
#pragma unroll
            for (int nt = 0; nt < 8; ++nt) {
                v2f b; b.x = bp[nt * 16];
                       b.y = bp[nt * 16 + C_DIM];
                oacc[0][nt] = __builtin_amdgcn_wmma_f32_16x16x4_f32(
                                  false, a0, false, b, (short)0, oacc[0][nt], false, false);
                oacc[1][nt] = __builtin_amdgcn_wmma_f32_16x16x4_f32(
                                  false, a1, false, b, (short)0, oacc[1][nt], false, false);
            }
        }
    }

    // ---- Epilogue: weighted scatter-add (exactly 2 adds/elem -> deterministic)
#pragma unroll
    for (int mt = 0; mt < 2; ++mt) {
#pragma unroll
        for (int j = 0; j < 8; ++j) {
            const int row = mt * 16 + 8 * lh + j;
            const float wgt = wS[row];
            if (wgt != 0.f) {
                float* obase = out + (size_t)tokS[row] * C_DIM + wv * 128 + l16;
#pragma unroll
                for (int nt = 0; nt < 8; ++nt)
                    atomicAdd(obase + nt * 16, oacc[mt][nt][j] * wgt);
            }
        }
    }
}

// ---------------------------------------------------------------------------
extern "C" void kernel_launch(void* const* d_in, const int* in_sizes, int n_in,
                              void* d_out, int out_size, void* d_ws, size_t ws_size,
                              hipStream_t stream)
{
    const float* x  = (const float*)d_in[0];   // [B,T,C]
    const float* rw = (const float*)d_in[1];   // [E,C]
    const float* w1 = (const float*)d_in[2];   // [E,C,H]
    const float* w2 = (const float*)d_in[3];   // [E,H,C]
    float* out = (float*)d_out;                // [B,T,C] = N_TOK*C_DIM floats

    char* ws    = (char*)d_ws;
    int*   cnt  = (int*)ws;                                        // 8 counters
    int*   ptok = (int*)(ws + 64);                                 // [E][N] tokens
    float* pw   = (float*)(ws + 64 + sizeof(int) * E_NUM * N_TOK); // [E][N] weights

    (void)hipMemsetAsync(cnt, 0, 64, stream);
    (void)hipMemsetAsync(out, 0, (size_t)out_size * sizeof(float), stream);

    moe_router<<<dim3(N_TOK / 8), dim3(256), 0, stream>>>(x, rw, cnt, ptok, pw);
    moe_expert_gemm<<<dim3(N_TOK / TM, E_NUM), dim3(256), 0, stream>>>(
        x, w1, w2, cnt, ptok, pw, out);
}